// GQA_60266981098012
// MI455X (gfx1250) — compile-verified
//
#include <hip/hip_runtime.h>
#include <hip/hip_bf16.h>
#include <math.h>

// Problem constants (match reference)
#define DIM  768
#define HQ   8
#define HKV  2
#define HD   64
#define SEQ  4096
#define QW   (HQ * HD)    // 512
#define KVW  (HKV * HD)   // 128

typedef __attribute__((ext_vector_type(2))) float v2f;
typedef __attribute__((ext_vector_type(8))) float v8f;

// D = A(16x4) * B(4x16) + C, fp32 WMMA (V_WMMA_F32_16X16X4_F32)
#define WMMA_F32(a, b, c) \
  __builtin_amdgcn_wmma_f32_16x16x4_f32(false, (a), false, (b), (short)0, (c), false, false)

// ---------------------------------------------------------------------------
// Generic wave-tiled GEMM: C[16x64 tile] = scale * (A @ B) + bias
// A row-major [M x K] (lda), B row-major [K x N] (ldb), C row-major (ldc).
// One wave (32 threads) per tile; 4 accumulators along N.
// Grid: (N/64, M/16). Requires K % 4 == 0, N % 64 == 0, M % 16 == 0.
// ---------------------------------------------------------------------------
__global__ __launch_bounds__(32) void gemm_nn_bias(
    const float* __restrict__ A, const float* __restrict__ B,
    const float* __restrict__ bias, float* __restrict__ C,
    int K, int lda, int ldb, int ldc, float scale)
{
  const int lane  = threadIdx.x & 31;
  const int lmod  = lane & 15;
  const int lhalf = lane >> 4;
  const int nBase = blockIdx.x * 64;
  const int mBase = blockIdx.y * 16;

  v8f acc0 = {}, acc1 = {}, acc2 = {}, acc3 = {};
  const float* ap = A + (long)(mBase + lmod) * lda + 2 * lhalf;
  const float* bp = B + (long)(2 * lhalf) * ldb + nBase + lmod;
  const long bstep = 4L * (long)ldb;

#pragma unroll 2
  for (int kk = 0; kk < K; kk += 4) {
    v2f a  = *(const v2f*)ap;
    v2f b0 = { bp[0],  bp[ldb] };
    v2f b1 = { bp[16], bp[16 + ldb] };
    v2f b2 = { bp[32], bp[32 + ldb] };
    v2f b3 = { bp[48], bp[48 + ldb] };
    acc0 = WMMA_F32(a, b0, acc0);
    acc1 = WMMA_F32(a, b1, acc1);
    acc2 = WMMA_F32(a, b2, acc2);
    acc3 = WMMA_F32(a, b3, acc3);
    ap += 4;
    bp += bstep;
  }

  const float bb0 = bias ? bias[nBase + lmod]      : 0.0f;
  const float bb1 = bias ? bias[nBase + 16 + lmod] : 0.0f;
  const float bb2 = bias ? bias[nBase + 32 + lmod] : 0.0f;
  const float bb3 = bias ? bias[nBase + 48 + lmod] : 0.0f;
  float* cp = C + (long)(mBase + 8 * lhalf) * ldc + nBase + lmod;
#pragma unroll
  for (int j = 0; j < 8; ++j) {
    float* r = cp + (long)j * ldc;
    r[0]  = acc0[j] * scale + bb0;
    r[16] = acc1[j] * scale + bb1;
    r[32] = acc2[j] * scale + bb2;
    r[48] = acc3[j] * scale + bb3;
  }
}

// ---------------------------------------------------------------------------
// Attention scores: att[h] = (q_h @ k_kv^T) / sqrt(HD)
// q layout [SEQ, QW] (head h at column h*HD), k layout [SEQ, KVW].
// A@B^T form: both fragments are contiguous float2 loads along d.
// Grid: (SEQ/64, SEQ/16, HQ), block 32.
// ---------------------------------------------------------------------------
__global__ __launch_bounds__(32) void attn_scores(
    const float* __restrict__ q, const float* __restrict__ kbuf,
    float* __restrict__ att)
{
  const int lane  = threadIdx.x & 31;
  const int lmod  = lane & 15;
  const int lhalf = lane >> 4;
  const int h     = blockIdx.z;
  const int kv    = h >> 2;            // n_rep = HQ/HKV = 4
  const int nBase = blockIdx.x * 64;
  const int mBase = blockIdx.y * 16;

  const float* ap = q    + (long)(mBase + lmod) * QW  + h  * HD + 2 * lhalf;
  const float* bp = kbuf + (long)(nBase + lmod) * KVW + kv * HD + 2 * lhalf;

  v8f acc0 = {}, acc1 = {}, acc2 = {}, acc3 = {};
#pragma unroll 4
  for (int kk = 0; kk < HD; kk += 4) {
    v2f a  = *(const v2f*)ap;
    v2f b0 = *(const v2f*)(bp);
    v2f b1 = *(const v2f*)(bp + 16 * KVW);
    v2f b2 = *(const v2f*)(bp + 32 * KVW);
    v2f b3 = *(const v2f*)(bp + 48 * KVW);
    acc0 = WMMA_F32(a, b0, acc0);
    acc1 = WMMA_F32(a, b1, acc1);
    acc2 = WMMA_F32(a, b2, acc2);
    acc3 = WMMA_F32(a, b3, acc3);
    ap += 4;
    bp += 4;
  }

  const float scale = 0.125f;          // 1/sqrt(64)
  float* cp = att + (long)h * SEQ * SEQ + (long)(mBase + 8 * lhalf) * SEQ + nBase + lmod;
#pragma unroll
  for (int j = 0; j < 8; ++j) {
    float* r = cp + (long)j * SEQ;
    r[0]  = acc0[j] * scale;
    r[16] = acc1[j] * scale;
    r[32] = acc2[j] * scale;
    r[48] = acc3[j] * scale;
  }
}

// ---------------------------------------------------------------------------
// Row softmax in place over att rows of length SEQ.
// Grid: (HQ*SEQ), block 256 (8 waves). 16 elements/thread held in registers.
// ---------------------------------------------------------------------------
__global__ __launch_bounds__(256) void softmax_rows(float* __restrict__ att)
{
  const long row = blockIdx.x;
  float* p = att + row * (long)SEQ;
  const int tid = threadIdx.x;

  float vals[16];
  float m = -INFINITY;
#pragma unroll
  for (int i = 0; i < 16; ++i) {
    vals[i] = p[tid + i * 256];
    m = fmaxf(m, vals[i]);
  }

  __shared__ float red[8];
#pragma unroll
  for (int off = 16; off > 0; off >>= 1)
    m = fmaxf(m, __shfl_xor(m, off, 32));
  if ((tid & 31) == 0) red[tid >> 5] = m;
  __syncthreads();
  m = red[0];
#pragma unroll
  for (int w = 1; w < 8; ++w) m = fmaxf(m, red[w]);
  __syncthreads();

  float s = 0.0f;
#pragma unroll
  for (int i = 0; i < 16; ++i) {
    vals[i] = __expf(vals[i] - m);
    s += vals[i];
  }
#pragma unroll
  for (int off = 16; off > 0; off >>= 1)
    s += __shfl_xor(s, off, 32);
  if ((tid & 31) == 0) red[tid >> 5] = s;
  __syncthreads();
  s = red[0];
#pragma unroll
  for (int w = 1; w < 8; ++w) s += red[w];

  const float inv = 1.0f / s;
#pragma unroll
  for (int i = 0; i < 16; ++i) p[tid + i * 256] = vals[i] * inv;
}

// ---------------------------------------------------------------------------
// ctx[:, h*HD:(h+1)*HD] = att[h] @ v_kv      (att: [SEQ,SEQ], v: [SEQ,KVW])
// Grid: (1, SEQ/16, HQ), block 32. N tile = 64 = HD.
// ---------------------------------------------------------------------------
__global__ __launch_bounds__(32) void attn_av(
    const float* __restrict__ att, const float* __restrict__ vbuf,
    float* __restrict__ ctx)
{
  const int lane  = threadIdx.x & 31;
  const int lmod  = lane & 15;
  const int lhalf = lane >> 4;
  const int h     = blockIdx.z;
  const int kv    = h >> 2;
  const int mBase = blockIdx.y * 16;

  const float* ap = att + (long)h * SEQ * SEQ + (long)(mBase + lmod) * SEQ + 2 * lhalf;
  const float* bp = vbuf + (long)(2 * lhalf) * KVW + kv * HD + lmod;

  v8f acc0 = {}, acc1 = {}, acc2 = {}, acc3 = {};
#pragma unroll 2
  for (int kk = 0; kk < SEQ; kk += 4) {
    v2f a  = *(const v2f*)ap;
    v2f b0 = { bp[0],  bp[KVW] };
    v2f b1 = { bp[16], bp[16 + KVW] };
    v2f b2 = { bp[32], bp[32 + KVW] };
    v2f b3 = { bp[48], bp[48 + KVW] };
    acc0 = WMMA_F32(a, b0, acc0);
    acc1 = WMMA_F32(a, b1, acc1);
    acc2 = WMMA_F32(a, b2, acc2);
    acc3 = WMMA_F32(a, b3, acc3);
    ap += 4;
    bp += 4L * KVW;
  }

  float* cp = ctx + (long)(mBase + 8 * lhalf) * QW + h * HD + lmod;
#pragma unroll
  for (int j = 0; j < 8; ++j) {
    float* r = cp + (long)j * QW;
    r[0]  = acc0[j];
    r[16] = acc1[j];
    r[32] = acc2[j];
    r[48] = acc3[j];
  }
}

// ---------------------------------------------------------------------------
extern "C" void kernel_launch(void* const* d_in, const int* in_sizes, int n_in,
                              void* d_out, int out_size, void* d_ws, size_t ws_size,
                              hipStream_t stream)
{
  (void)in_sizes; (void)n_in; (void)out_size; (void)ws_size;

  const float* x  = (const float*)d_in[0];
  const float* Wq = (const float*)d_in[1];
  const float* bq = (const float*)d_in[2];
  const float* Wk = (const float*)d_in[3];
  const float* bk = (const float*)d_in[4];
  const float* Wv = (const float*)d_in[5];
  const float* bv = (const float*)d_in[6];
  const float* Wo = (const float*)d_in[7];
  const float* bo = (const float*)d_in[8];

  float* out = (float*)d_out;                 // [SEQ, DIM]
  float* att = out + (long)SEQ * DIM;         // [HQ, SEQ, SEQ]

  float* q   = (float*)d_ws;                  // [SEQ, QW]
  float* k   = q + (long)SEQ * QW;            // [SEQ, KVW]
  float* v   = k + (long)SEQ * KVW;           // [SEQ, KVW]
  float* ctx = v + (long)SEQ * KVW;           // [SEQ, QW]

  const dim3 wave(32);

  // Q/K/V projections: x[SEQ,DIM] @ W + b
  gemm_nn_bias<<<dim3(QW / 64,  SEQ / 16), wave, 0, stream>>>(x, Wq, bq, q, DIM, DIM, QW,  QW,  1.0f);
  gemm_nn_bias<<<dim3(KVW / 64, SEQ / 16), wave, 0, stream>>>(x, Wk, bk, k, DIM, DIM, KVW, KVW, 1.0f);
  gemm_nn_bias<<<dim3(KVW / 64, SEQ / 16), wave, 0, stream>>>(x, Wv, bv, v, DIM, DIM, KVW, KVW, 1.0f);

  // Scores (scaled), softmax, att @ V
  attn_scores<<<dim3(SEQ / 64, SEQ / 16, HQ), wave, 0, stream>>>(q, k, att);
  softmax_rows<<<dim3(HQ * SEQ), dim3(256), 0, stream>>>(att);
  attn_av<<<dim3(1, SEQ / 16, HQ), wave, 0, stream>>>(att, v, ctx);

  // Output projection: ctx[SEQ,QW] @ Wo[QW,DIM] + bo
  gemm_nn_bias<<<dim3(DIM / 64, SEQ / 16), wave, 0, stream>>>(ctx, Wo, bo, out, QW, QW, DIM, DIM, 1.0f);
}